// GCNNodeClassifier_50766513439532
// MI455X (gfx1250) — compile-verified
//
#include <hip/hip_runtime.h>
#include <hip/hip_bf16.h>

typedef __attribute__((ext_vector_type(2))) float v2f;
typedef __attribute__((ext_vector_type(8))) float v8f;

#define N_NODES 50000
#define D_IN    512
#define D_HID   2048
#define N_CLS   40
#define N_EDGES 200000

#define BM 128
#define BN 128
#define BK 32
#define LDK (BK + 2)   // k-stride in LDS: keeps v2f 8B-aligned, staggers banks

// ------------------------------------------------------------------
// degree / normalization
// ------------------------------------------------------------------
__global__ void deg_count_kernel(const int* __restrict__ dst,
                                 int* __restrict__ degI, int e) {
    int t = blockIdx.x * blockDim.x + threadIdx.x;
    if (t < e) atomicAdd(&degI[dst[t]], 1);
}

__global__ void dinv_kernel(const int* __restrict__ degI,
                            float* __restrict__ dinv, int n) {
    int t = blockIdx.x * blockDim.x + threadIdx.x;
    if (t < n) {
        // +1 accounts for the self loop; degree is always >= 1
        dinv[t] = rsqrtf((float)(degI[t] + 1));
    }
}

// ------------------------------------------------------------------
// fp32 WMMA GEMM: C[M,N] = A[M,K] @ B[K,N]   (row-major, K % 32 == 0)
// block = 256 threads (8 waves), block tile 128x128, K-step 32.
// Wave w computes rows [w*16, w*16+16) x 128 cols via 8 accumulator tiles.
// LDS tiles are stored k-contiguous so every WMMA fragment is ONE
// 8-byte-aligned v2f -> single ds_load_b64 into an even VGPR pair.
// ------------------------------------------------------------------
__global__ __launch_bounds__(256)
void wmma_gemm_f32(const float* __restrict__ A, const float* __restrict__ B,
                   float* __restrict__ C, int M, int N, int K) {
    __shared__ float As[BM][LDK];   // As[m][k]
    __shared__ float Bs[BN][LDK];   // Bs[n][k]  (transposed B tile)

    const int tid   = threadIdx.x;
    const int lane  = tid & 31;
    const int wave  = tid >> 5;        // 0..7
    const int h     = lane >> 4;       // half of wave32
    const int l16   = lane & 15;
    const int blockM = blockIdx.y * BM;
    const int blockN = blockIdx.x * BN;
    const int mLocal = wave * 16 + l16;

    v8f acc[8];
    const v8f vzero = {0.f, 0.f, 0.f, 0.f, 0.f, 0.f, 0.f, 0.f};
#pragma unroll
    for (int nt = 0; nt < 8; ++nt) acc[nt] = vzero;

    for (int kb = 0; kb < K; kb += BK) {
        // ---- A tile: 128x32 floats, coalesced float4 reads, k-contiguous store
#pragma unroll
        for (int i = 0; i < 4; ++i) {
            int idx  = tid + i * 256;        // 0..1023 float4 slots
            int row  = idx >> 3;             // 0..127
            int c4   = (idx & 7) << 2;       // 0,4,..,28
            int gRow = blockM + row;
            float4 v = make_float4(0.f, 0.f, 0.f, 0.f);
            if (gRow < M) {
                const float* ap = A + (size_t)gRow * K + kb + c4;
                v = *(const float4*)ap;
                if (kb + BK < K) {
                    __builtin_prefetch(ap + BK, 0, 1);   // global_prefetch_b8
                }
            }
            As[row][c4 + 0] = v.x;
            As[row][c4 + 1] = v.y;
            As[row][c4 + 2] = v.z;
            As[row][c4 + 3] = v.w;
        }
        // ---- B tile: 32x128 floats, coalesced along n, transposed into Bs[n][k]
#pragma unroll
        for (int i = 0; i < 4; ++i) {
            int idx  = tid + i * 256;        // 0..1023 float4 slots
            int krow = idx >> 5;             // 0..31
            int c4   = (idx & 31) << 2;      // 0,4,..,124
            int gk   = kb + krow;
            const float* bp = B + (size_t)gk * N + blockN + c4;
#pragma unroll
            for (int j = 0; j < 4; ++j) {
                int gn = blockN + c4 + j;
                Bs[c4 + j][krow] = (gn < N) ? bp[j] : 0.f;
            }
        }
        __syncthreads();

#pragma unroll
        for (int kk = 0; kk < BK; kk += 4) {
            v2f a = *(const v2f*)&As[mLocal][kk + 2 * h];
#pragma unroll
            for (int nt = 0; nt < 8; ++nt) {
                v2f b = *(const v2f*)&Bs[nt * 16 + l16][kk + 2 * h];
                acc[nt] = __builtin_amdgcn_wmma_f32_16x16x4_f32(
                    false, a, false, b, (short)0, acc[nt], false, false);
            }
        }
        __syncthreads();
    }

    // ---- store: C/D layout -> lane row = r + 8h, col = l16
#pragma unroll
    for (int nt = 0; nt < 8; ++nt) {
        int gCol = blockN + nt * 16 + l16;
        if (gCol >= N) continue;
#pragma unroll
        for (int r = 0; r < 8; ++r) {
            int gRow = blockM + wave * 16 + r + 8 * h;
            if (gRow < M) C[(size_t)gRow * N + gCol] = acc[nt][r];
        }
    }
}

// ------------------------------------------------------------------
// graph aggregation: out[dst] += h[src] * dinv[src]*dinv[dst]
// ------------------------------------------------------------------
__global__ void edge_agg_kernel(const float* __restrict__ h,
                                const int* __restrict__ src,
                                const int* __restrict__ dst,
                                const float* __restrict__ dinv,
                                float* __restrict__ out, int e, int d) {
    for (int idx = blockIdx.x; idx < e; idx += gridDim.x) {
        int s = src[idx];
        int t = dst[idx];
        float nrm = dinv[s] * dinv[t];
        const float* hs = h + (size_t)s * d;
        float* od = out + (size_t)t * d;
        for (int f = threadIdx.x; f < d; f += blockDim.x) {
            unsafeAtomicAdd(&od[f], hs[f] * nrm);
        }
    }
}

// self loops: out[i] += h[i] * dinv[i]^2  (exclusive per element, no atomics)
__global__ void self_loop_kernel(const float* __restrict__ h,
                                 const float* __restrict__ dinv,
                                 float* __restrict__ out,
                                 long long total, int d) {
    long long stride = (long long)gridDim.x * blockDim.x;
    for (long long t = (long long)blockIdx.x * blockDim.x + threadIdx.x;
         t < total; t += stride) {
        int i = (int)(t / d);
        float w = dinv[i];
        out[t] += h[t] * w * w;
    }
}

// ------------------------------------------------------------------
// PairNorm: stats pass (colsum + total sum of squares of x+bias)
// grid = (d/256, chunks); requires d % 256 == 0
// ------------------------------------------------------------------
__global__ __launch_bounds__(256)
void pairnorm_stats_kernel(const float* __restrict__ x,
                           const float* __restrict__ bias,
                           float* __restrict__ colsum,
                           float* __restrict__ sumsq,
                           int n, int d, int rowsPerChunk) {
    int col = blockIdx.x * blockDim.x + threadIdx.x;
    int r0 = blockIdx.y * rowsPerChunk;
    int r1 = min(n, r0 + rowsPerChunk);
    float b = bias[col];
    float s = 0.f, ss = 0.f;
    for (int r = r0; r < r1; ++r) {
        float v = x[(size_t)r * d + col] + b;
        s += v;
        ss += v * v;
    }
    unsafeAtomicAdd(&colsum[col], s);
    __shared__ float red[256];
    red[threadIdx.x] = ss;
    __syncthreads();
    for (int o = 128; o > 0; o >>= 1) {
        if (threadIdx.x < o) red[threadIdx.x] += red[threadIdx.x + o];
        __syncthreads();
    }
    if (threadIdx.x == 0) unsafeAtomicAdd(sumsq, red[0]);
}

// sum_i ||x_i - mu||^2 = sumsq_total - n * sum_d mu_d^2
__global__ __launch_bounds__(256)
void pairnorm_finalize_kernel(const float* __restrict__ colsum,
                              const float* __restrict__ sumsq,
                              float* __restrict__ denomInv, int n, int d) {
    float invN = 1.f / (float)n;
    float p = 0.f;
    for (int c = threadIdx.x; c < d; c += 256) {
        float mu = colsum[c] * invN;
        p += mu * mu;
    }
    __shared__ float red[256];
    red[threadIdx.x] = p;
    __syncthreads();
    for (int o = 128; o > 0; o >>= 1) {
        if (threadIdx.x < o) red[threadIdx.x] += red[threadIdx.x + o];
        __syncthreads();
    }
    if (threadIdx.x == 0) {
        float sc = sumsq[0] - (float)n * red[0];   // centered total sumsq
        denomInv[0] = rsqrtf(1e-6f + sc * invN);
    }
}

__global__ void pairnorm_relu_kernel(const float* __restrict__ x,
                                     const float* __restrict__ bias,
                                     const float* __restrict__ colsum,
                                     const float* __restrict__ denomInv,
                                     float* __restrict__ out,
                                     long long total, int d, float invN) {
    float di = denomInv[0];
    long long stride = (long long)gridDim.x * blockDim.x;
    for (long long t = (long long)blockIdx.x * blockDim.x + threadIdx.x;
         t < total; t += stride) {
        int c = (int)(t % d);
        float mu = colsum[c] * invN;
        float v = (x[t] + bias[c] - mu) * di;
        out[t] = fmaxf(v, 0.f);
    }
}

__global__ void bias_add_kernel(float* __restrict__ out,
                                const float* __restrict__ b,
                                long long total, int d) {
    long long stride = (long long)gridDim.x * blockDim.x;
    for (long long t = (long long)blockIdx.x * blockDim.x + threadIdx.x;
         t < total; t += stride) {
        out[t] += b[(int)(t % d)];
    }
}

// ------------------------------------------------------------------
// launch
// ------------------------------------------------------------------
extern "C" void kernel_launch(void* const* d_in, const int* in_sizes, int n_in,
                              void* d_out, int out_size, void* d_ws, size_t ws_size,
                              hipStream_t stream) {
    const float* x  = (const float*)d_in[0];
    const int*   ei = (const int*)d_in[1];
    const float* W1 = (const float*)d_in[2];
    const float* b1 = (const float*)d_in[3];
    const float* W2 = (const float*)d_in[4];
    const float* b2 = (const float*)d_in[5];
    const float* W3 = (const float*)d_in[6];
    const float* b3 = (const float*)d_in[7];
    float* out = (float*)d_out;

    // workspace layout
    float* bufA   = (float*)d_ws;                          // 50000*2048
    float* bufB   = bufA + (size_t)N_NODES * D_HID;        // 50000*2048
    float* bufC   = bufB + (size_t)N_NODES * D_HID;        // 50000*40
    int*   degI   = (int*)(bufC + (size_t)N_NODES * N_CLS);
    float* dinv   = (float*)(degI + N_NODES);
    float* colsum = dinv + N_NODES;                        // D_HID floats
    float* scalars = colsum + D_HID;                       // [0]=denomInv, [1]=sumsq

    const int* srcIdx = ei;
    const int* dstIdx = ei + N_EDGES;

    const long long totalHid = (long long)N_NODES * D_HID;
    const long long totalCls = (long long)N_NODES * N_CLS;
    const int ewBlocksHid = (int)((totalHid + 255) / 256);
    const int ewBlocksCls = (int)((totalCls + 255) / 256);
    const int mTiles = (N_NODES + BM - 1) / BM;            // 391

    // ---- degrees -> dinv
    hipMemsetAsync(degI, 0, N_NODES * sizeof(int), stream);
    deg_count_kernel<<<(N_EDGES + 255) / 256, 256, 0, stream>>>(dstIdx, degI, N_EDGES);
    dinv_kernel<<<(N_NODES + 255) / 256, 256, 0, stream>>>(degI, dinv, N_NODES);

    const int chunks = 128;
    const int rowsPerChunk = (N_NODES + chunks - 1) / chunks;
    const float invN = 1.f / (float)N_NODES;

    // ================= layer 1 =================
    wmma_gemm_f32<<<dim3(D_HID / BN, mTiles), 256, 0, stream>>>(
        x, W1, bufA, N_NODES, D_HID, D_IN);
    hipMemsetAsync(bufB, 0, (size_t)totalHid * sizeof(float), stream);
    edge_agg_kernel<<<N_EDGES, 256, 0, stream>>>(bufA, srcIdx, dstIdx, dinv,
                                                 bufB, N_EDGES, D_HID);
    self_loop_kernel<<<ewBlocksHid, 256, 0, stream>>>(bufA, dinv, bufB, totalHid, D_HID);
    hipMemsetAsync(colsum, 0, (D_HID + 4) * sizeof(float), stream);
    pairnorm_stats_kernel<<<dim3(D_HID / 256, chunks), 256, 0, stream>>>(
        bufB, b1, colsum, scalars + 1, N_NODES, D_HID, rowsPerChunk);
    pairnorm_finalize_kernel<<<1, 256, 0, stream>>>(colsum, scalars + 1, scalars,
                                                    N_NODES, D_HID);
    pairnorm_relu_kernel<<<ewBlocksHid, 256, 0, stream>>>(
        bufB, b1, colsum, scalars, bufA, totalHid, D_HID, invN);

    // ================= layer 2 =================
    wmma_gemm_f32<<<dim3(D_HID / BN, mTiles), 256, 0, stream>>>(
        bufA, W2, bufB, N_NODES, D_HID, D_HID);
    hipMemsetAsync(bufA, 0, (size_t)totalHid * sizeof(float), stream);
    edge_agg_kernel<<<N_EDGES, 256, 0, stream>>>(bufB, srcIdx, dstIdx, dinv,
                                                 bufA, N_EDGES, D_HID);
    self_loop_kernel<<<ewBlocksHid, 256, 0, stream>>>(bufB, dinv, bufA, totalHid, D_HID);
    hipMemsetAsync(colsum, 0, (D_HID + 4) * sizeof(float), stream);
    pairnorm_stats_kernel<<<dim3(D_HID / 256, chunks), 256, 0, stream>>>(
        bufA, b2, colsum, scalars + 1, N_NODES, D_HID, rowsPerChunk);
    pairnorm_finalize_kernel<<<1, 256, 0, stream>>>(colsum, scalars + 1, scalars,
                                                    N_NODES, D_HID);
    pairnorm_relu_kernel<<<ewBlocksHid, 256, 0, stream>>>(
        bufA, b2, colsum, scalars, bufB, totalHid, D_HID, invN);

    // ================= layer 3 =================
    wmma_gemm_f32<<<dim3(1, mTiles), 256, 0, stream>>>(
        bufB, W3, bufC, N_NODES, N_CLS, D_HID);
    hipMemsetAsync(out, 0, (size_t)totalCls * sizeof(float), stream);
    edge_agg_kernel<<<N_EDGES, 64, 0, stream>>>(bufC, srcIdx, dstIdx, dinv,
                                                out, N_EDGES, N_CLS);
    self_loop_kernel<<<ewBlocksCls, 256, 0, stream>>>(bufC, dinv, out, totalCls, N_CLS);
    bias_add_kernel<<<ewBlocksCls, 256, 0, stream>>>(out, b3, totalCls, N_CLS);
}